// GraphCrossModule_86569360818432
// MI455X (gfx1250) — compile-verified
//
#include <hip/hip_runtime.h>
#include <hip/hip_bf16.h>

#define TPB 256

typedef __attribute__((ext_vector_type(2))) float f32x2;
typedef __attribute__((ext_vector_type(8))) float f32x8;

__device__ __forceinline__ void atomicAddF32(float* p, float v) {
    __hip_atomic_fetch_add(p, v, __ATOMIC_RELAXED, __HIP_MEMORY_SCOPE_AGENT);
}

// ---------------------------------------------------------------------------
// GEMM: Y[n x 128] = X[n x K] @ W[K x 128]   (fp32 WMMA 16x16x4)
// One block = 8 waves; block owns 16 rows, wave w owns cols [16w,16w+16).
// ---------------------------------------------------------------------------
__global__ __launch_bounds__(TPB) void k_gemm_wmma(
    const float* __restrict__ X, const float* __restrict__ W,
    float* __restrict__ Y, int n, int K)
{
    extern __shared__ float Xs[]; // 16 * K floats
    const int tid  = threadIdx.x;
    const int row0 = blockIdx.x * 16;

    for (int idx = tid; idx < 16 * K; idx += TPB) {
        int r = idx / K, k = idx - r * K;
        int gr = row0 + r;
        Xs[idx] = (gr < n) ? X[(size_t)gr * K + k] : 0.0f;
    }
    __syncthreads();

    const int wave  = tid >> 5;
    const int lane  = tid & 31;
    const int m     = lane & 15;
    const int khalf = (lane >> 4) << 1;      // 0 or 2
    const int col   = (wave << 4) + m;

    f32x8 acc = {0.f,0.f,0.f,0.f,0.f,0.f,0.f,0.f};
    for (int kk = 0; kk < K; kk += 4) {
        f32x2 a, b;
        a.x = Xs[m * K + kk + khalf];
        a.y = Xs[m * K + kk + khalf + 1];
        b.x = W[(size_t)(kk + khalf)     * 128 + col];
        b.y = W[(size_t)(kk + khalf + 1) * 128 + col];
        acc = __builtin_amdgcn_wmma_f32_16x16x4_f32(
            false, a, false, b, (short)0, acc, false, false);
    }
#pragma unroll
    for (int v = 0; v < 8; ++v) {
        int r = row0 + v + ((lane >> 4) << 3);
        if (r < n) Y[(size_t)r * 128 + col] = acc[v];
    }
}

// ---------------------------------------------------------------------------
// Graph / elementwise kernels
// ---------------------------------------------------------------------------
__global__ void k_zero_f(float* p, long long n) {
    long long i = (long long)blockIdx.x * TPB + threadIdx.x;
    if (i < n) p[i] = 0.0f;
}
__global__ void k_fill_bias(float* out, const float* __restrict__ b, long long n128) {
    long long i = (long long)blockIdx.x * TPB + threadIdx.x;
    if (i < n128) out[i] = b[i & 127];
}
__global__ void k_relu(float* p, long long n) {
    long long i = (long long)blockIdx.x * TPB + threadIdx.x;
    if (i < n) p[i] = fmaxf(p[i], 0.0f);
}
__global__ void k_prelu(float* p, long long n, const float* __restrict__ aptr, int aidx) {
    long long i = (long long)blockIdx.x * TPB + threadIdx.x;
    if (i >= n) return;
    float a = aptr[aidx];
    float v = p[i];
    p[i] = (v >= 0.0f) ? v : a * v;
}
__global__ void k_sigmoid(float* out, const float* __restrict__ in, int n) {
    int i = blockIdx.x * TPB + threadIdx.x;
    if (i < n) out[i] = 1.0f / (1.0f + expf(-in[i]));
}
__global__ void k_deg(float* deg, const int* __restrict__ dst,
                      const float* __restrict__ ew, int nE) {
    int e = blockIdx.x * TPB + threadIdx.x;
    if (e >= nE) return;
    atomicAddF32(&deg[dst[e]], ew ? ew[e] : 1.0f);
}
__global__ void k_isqt(float* p, int n) {
    int i = blockIdx.x * TPB + threadIdx.x;
    if (i >= n) return;
    float d = p[i];
    p[i] = (d > 0.0f) ? rsqrtf(fmaxf(d, 1.0f)) : 0.0f;
}
// out[dst] += x[src] * (ew * isq[src] * isq[dst]); 8 edges/block, wave per edge
__global__ __launch_bounds__(TPB) void k_agg(
    float* __restrict__ out, const float* __restrict__ x,
    const int* __restrict__ src, const int* __restrict__ dst,
    const float* __restrict__ ew, const float* __restrict__ isq, int nE)
{
    int e = blockIdx.x * 8 + (threadIdx.x >> 5);
    if (e >= nE) return;
    int lane = threadIdx.x & 31;
    int s = src[e], d = dst[e];
    float w = ew ? ew[e] : 1.0f;
    float coef = w * isq[s] * isq[d];
    if (coef == 0.0f) return;
    const float4* xin = (const float4*)(x + (size_t)s * 128);
    float4 v = xin[lane];
    float* o = out + (size_t)d * 128 + lane * 4;
    atomicAddF32(o + 0, v.x * coef);
    atomicAddF32(o + 1, v.y * coef);
    atomicAddF32(o + 2, v.z * coef);
    atomicAddF32(o + 3, v.w * coef);
}
__global__ void k_rowdot(float* out, const float* __restrict__ a,
                         const float* __restrict__ b, int n) {
    int i = blockIdx.x * TPB + threadIdx.x;
    if (i >= n) return;
    const float4* pa = (const float4*)(a + (size_t)i * 128);
    const float4* pb = (const float4*)(b + (size_t)i * 128);
    float s = 0.0f;
    for (int t = 0; t < 32; ++t) {
        float4 va = pa[t], vb = pb[t];
        s += va.x * vb.x + va.y * vb.y + va.z * vb.z + va.w * vb.w;
    }
    out[i] = s;
}
__global__ void k_rowscale(float* out, const float* __restrict__ x,
                           const float* __restrict__ s, long long n128) {
    long long i = (long long)blockIdx.x * TPB + threadIdx.x;
    if (i < n128) out[i] = x[i] * s[i >> 7];
}
__global__ void k_gather_rows(float* out, const float* __restrict__ x,
                              const int* __restrict__ idx, long long n128) {
    long long i = (long long)blockIdx.x * TPB + threadIdx.x;
    if (i < n128) out[i] = x[(size_t)idx[i >> 7] * 128 + (i & 127)];
}
__global__ void k_gather_scale(float* out, const float* __restrict__ x,
                               const float* __restrict__ s,
                               const int* __restrict__ sel, long long k128) {
    long long i = (long long)blockIdx.x * TPB + threadIdx.x;
    if (i >= k128) return;
    int sr = sel[i >> 7];
    out[i] = x[(size_t)sr * 128 + (i & 127)] * s[sr];
}
__global__ void k_scatter_rows(float* out, const float* __restrict__ xs,
                               const int* __restrict__ sel, long long k128) {
    long long i = (long long)blockIdx.x * TPB + threadIdx.x;
    if (i >= k128) return;
    out[(size_t)sel[i >> 7] * 128 + (i & 127)] = xs[i];
}
__global__ void k_fill_int(int* p, int n, int val) {
    int i = blockIdx.x * TPB + threadIdx.x;
    if (i < n) p[i] = val;
}
__global__ void k_set_remap(int* remap, const int* __restrict__ sel, int k) {
    int i = blockIdx.x * TPB + threadIdx.x;
    if (i < k) remap[sel[i]] = i;
}
__global__ void k_pool_edges(int* s2, int* d2, float* w2,
                             const int* __restrict__ s, const int* __restrict__ d,
                             const float* __restrict__ w,
                             const int* __restrict__ remap, int k, int nE) {
    int e = blockIdx.x * TPB + threadIdx.x;
    if (e >= nE) return;
    int ns = remap[s[e]], nd = remap[d[e]];
    bool keep = (ns < k) && (nd < k);
    s2[e] = keep ? ns : 0;
    d2[e] = keep ? nd : 0;
    w2[e] = keep ? (w ? w[e] : 1.0f) : 0.0f;
}
// stable argsort-by-descending-score via rank counting (matches jnp.argsort on -scores)
__global__ void k_rank_topk(int* sel, const float* __restrict__ sc, int n, int k) {
    int i = blockIdx.x * TPB + threadIdx.x;
    if (i >= n) return;
    float si = sc[i];
    int r = 0;
    for (int j = 0; j < n; ++j) {
        float sj = sc[j];
        r += (sj > si) || (sj == si && j < i);
    }
    if (r < k) sel[r] = i;
}
__global__ void k_add_wsum(float* out, const float* __restrict__ a,
                           const float* __restrict__ b, const float* __restrict__ c,
                           float sb, float sc, long long n) {
    long long i = (long long)blockIdx.x * TPB + threadIdx.x;
    if (i >= n) return;
    float v = a[i] + sb * b[i];
    if (c) v += sc * c[i];
    out[i] = v;
}
__global__ void k_concat(float* out, const float* __restrict__ a,
                         const float* __restrict__ b, long long n256) {
    long long i = (long long)blockIdx.x * TPB + threadIdx.x;
    if (i >= n256) return;
    long long r = i >> 8;
    int dcol = (int)(i & 255);
    out[i] = (dcol < 128) ? a[r * 128 + dcol] : b[r * 128 + dcol - 128];
}

// ---------------------------------------------------------------------------
// Host orchestration
// ---------------------------------------------------------------------------
extern "C" void kernel_launch(void* const* d_in, const int* in_sizes, int n_in,
                              void* d_out, int out_size, void* d_ws, size_t ws_size,
                              hipStream_t stream)
{
    const float* feat     = (const float*)d_in[0];
    const int*   src1     = (const int*)d_in[1];
    const int*   dst1     = (const int*)d_in[2];
    const int*   perm1    = (const int*)d_in[3];
    const int*   perm2    = (const int*)d_in[4];
    const float* W_start1 = (const float*)d_in[5];
    const float* b_start1 = (const float*)d_in[6];
    const float* W_start2 = (const float*)d_in[7];
    const float* b_start2 = (const float*)d_in[8];
    const float* W_end    = (const float*)d_in[9];
    const float* b_end    = (const float*)d_in[10];
    const float* Wphi     = (const float*)d_in[11];
    const float* Wpsi     = (const float*)d_in[12];
    const float* Wdisc    = (const float*)d_in[13];
    const float* is_a     = (const float*)d_in[14];
    const float* Wm       = (const float*)d_in[15];
    const float* bm       = (const float*)d_in[16];
    const float* Wp       = (const float*)d_in[17];
    const float* bp       = (const float*)d_in[18];
    const float* Wu       = (const float*)d_in[19];
    const float* bu       = (const float*)d_in[20];

    const int N  = in_sizes[3];      // perm1 length
    const int E2 = in_sizes[1];      // concatenated edge count
    const int K1 = in_sizes[4];      // perm2 length = ceil(0.8*N)
    const int K2 = (4 * K1 + 4) / 5; // ceil(0.8*K1)
    const int H  = 128;
    const size_t HH = (size_t)H * H;

    // --- workspace bump allocator ---
    char* wsp = (char*)d_ws;
    size_t off = 0;
    auto allocB = [&](size_t bytes) -> void* {
        void* p = wsp + off;
        off += (bytes + 255) & ~(size_t)255;
        return p;
    };
    auto allocF = [&](size_t cnt) -> float* { return (float*)allocB(cnt * 4); };
    auto allocI = [&](size_t cnt) -> int*   { return (int*)allocB(cnt * 4); };

    float* x1    = allocF((size_t)N * H);
    float* feat0 = allocF((size_t)N * H);
    float* down1 = allocF((size_t)N * H);
    float* t1a   = allocF((size_t)N * H);
    float* t1b   = allocF((size_t)N * H);
    float* t1c   = allocF((size_t)N * H);
    float* s2buf = allocF((size_t)N * H);         // f21 / s2_out
    float* catb  = allocF((size_t)N * 2 * H);

    float* x2    = allocF((size_t)K1 * H);
    float* res2  = allocF((size_t)K1 * H);
    float* down2 = allocF((size_t)K1 * H);
    float* t2a   = allocF((size_t)K1 * H);
    float* t2b   = allocF((size_t)K1 * H);
    float* t2c   = allocF((size_t)K1 * H);
    float* f12   = allocF((size_t)K1 * H);
    float* f32b  = allocF((size_t)K1 * H);
    float* s3buf = allocF((size_t)K1 * H);

    float* x3    = allocF((size_t)K2 * H);
    float* res3  = allocF((size_t)K2 * H);
    float* t3a   = allocF((size_t)K2 * H);
    float* t3b   = allocF((size_t)K2 * H);
    float* f23   = allocF((size_t)K2 * H);

    float* scores1 = allocF(N);
    float* scores2 = allocF(K1);
    float* isq1    = allocF(N);
    float* isq2    = allocF(K1);
    float* isq3    = allocF(K2);
    float* w2      = allocF(E2);
    float* w3      = allocF(E2);

    int* sel1  = allocI(K1);
    int* sel2  = allocI(K2);
    int* remap = allocI(N);
    int* src2  = allocI(E2);
    int* dst2  = allocI(E2);
    int* src3  = allocI(E2);
    int* dst3  = allocI(E2);

    // --- outputs ---
    float* out_x = (float*)d_out;                 // N*128
    float* pos1  = out_x + (size_t)N * H;         // logit1 row 0
    float* neg1  = pos1 + N;                      // logit1 row 1
    float* pos2  = neg1 + N;                      // logit2 row 0
    float* neg2  = pos2 + K1;                     // logit2 row 1

    // --- launch helpers ---
    auto gridL = [](long long n) { return dim3((unsigned)((n + TPB - 1) / TPB)); };
    auto gemm = [&](float* Y, const float* X, const float* W, int n, int K) {
        dim3 g((n + 15) / 16);
        k_gemm_wmma<<<g, TPB, (size_t)16 * K * sizeof(float), stream>>>(X, W, Y, n, K);
    };
    auto zerof = [&](float* p, long long n) { k_zero_f<<<gridL(n), TPB, 0, stream>>>(p, n); };
    auto fill_bias = [&](float* o, const float* b, int n) {
        long long c = (long long)n * H;
        k_fill_bias<<<gridL(c), TPB, 0, stream>>>(o, b, c);
    };
    struct G { const int* s; const int* d; const float* w; const float* isq; int nE; int n; };
    auto agg = [&](float* o, const float* x, const G& g) {
        k_agg<<<dim3((g.nE + 7) / 8), TPB, 0, stream>>>(o, x, g.s, g.d, g.w, g.isq, g.nE);
    };
    auto relu = [&](float* p, long long n) { k_relu<<<gridL(n), TPB, 0, stream>>>(p, n); };
    auto prelu = [&](float* p, long long n, int lvl) {
        k_prelu<<<gridL(n), TPB, 0, stream>>>(p, n, is_a, lvl);
    };
    auto copyF = [&](float* dst, const float* src, size_t cnt) {
        hipMemcpyAsync(dst, src, cnt * sizeof(float), hipMemcpyDeviceToDevice, stream);
    };
    // gcn: out = normagg(x @ W) + b   (bias folded as init value of accumulator)
    auto gcn_to = [&](float* o, float* tmp, const float* x, const float* W,
                      const float* b, const G& g) {
        gemm(tmp, x, W, g.n, H);
        fill_bias(o, b, g.n);
        agg(o, tmp, g);
    };
    auto build_graph_norm = [&](float* isq, const int* d, const float* w, int nE, int n) {
        zerof(isq, n);
        k_deg<<<gridL(nE), TPB, 0, stream>>>(isq, d, w, nE);
        k_isqt<<<gridL(n), TPB, 0, stream>>>(isq, n);
    };
    auto index_select = [&](const float* x, const int* perm, int lvl, const G& g,
                            float* ta, float* tb, float* tc,
                            float* pos, float* neg, float* scores, float* down,
                            int* sel, int k) {
        const float* Aphi = Wphi  + (size_t)lvl * HH;
        const float* Apsi = Wpsi  + (size_t)lvl * HH;
        const float* Adis = Wdisc + (size_t)lvl * HH;
        const int n = g.n;
        const long long nH = (long long)n * H;
        gemm(ta, x, Aphi, n, H); prelu(ta, nH, lvl);               // h_self
        zerof(tb, nH); agg(tb, x, g);                              // normagg(x)
        gemm(tc, tb, Apsi, n, H); prelu(tc, nH, lvl);              // h_neigh
        gemm(tb, tc, Adis, n, H);                                  // proj
        k_rowdot<<<gridL(n), TPB, 0, stream>>>(pos, ta, tb, n);
        k_gather_rows<<<gridL(nH), TPB, 0, stream>>>(tc, x, perm, nH);  // x_neg
        gemm(ta, tc, Aphi, n, H); prelu(ta, nH, lvl);
        k_rowdot<<<gridL(n), TPB, 0, stream>>>(neg, ta, tb, n);
        k_sigmoid<<<gridL(n), TPB, 0, stream>>>(scores, pos, n);
        k_rowscale<<<gridL(nH), TPB, 0, stream>>>(down, x, scores, nH);
        k_rank_topk<<<gridL(n), TPB, 0, stream>>>(sel, scores, n, k);
    };

    // ===================== forward =====================
    build_graph_norm(isq1, dst1, nullptr, E2, N);
    G g1{src1, dst1, nullptr, isq1, E2, N};

    // x1 = gcn(feat, W_start1)
    gcn_to(x1, t1a, feat, W_start1, b_start1, g1);
    copyF(feat0, x1, (size_t)N * H);

    // index_select level 1
    index_select(x1, perm1, 0, g1, t1a, t1b, t1c, pos1, neg1, scores1, down1, sel1, K1);

    // pool -> level 2 graph
    k_gather_scale<<<gridL((long long)K1 * H), TPB, 0, stream>>>(x2, x1, scores1, sel1, (long long)K1 * H);
    k_fill_int<<<gridL(N), TPB, 0, stream>>>(remap, N, K1);
    k_set_remap<<<gridL(K1), TPB, 0, stream>>>(remap, sel1, K1);
    k_pool_edges<<<gridL(E2), TPB, 0, stream>>>(src2, dst2, w2, src1, dst1, nullptr, remap, K1, E2);
    build_graph_norm(isq2, dst2, w2, E2, K1);
    G g2{src2, dst2, w2, isq2, E2, K1};

    // x2 = gcn(x2, W_start2)
    gcn_to(t2b, t2a, x2, W_start2, b_start2, g2);
    copyF(x2, t2b, (size_t)K1 * H);

    // index_select level 2
    index_select(x2, perm2, 1, g2, t2a, t2b, t2c, pos2, neg2, scores2, down2, sel2, K2);

    // pool -> level 3 graph
    k_gather_scale<<<gridL((long long)K2 * H), TPB, 0, stream>>>(x3, x2, scores2, sel2, (long long)K2 * H);
    k_fill_int<<<gridL(K1), TPB, 0, stream>>>(remap, K1, K2);
    k_set_remap<<<gridL(K2), TPB, 0, stream>>>(remap, sel2, K2);
    k_pool_edges<<<gridL(E2), TPB, 0, stream>>>(src3, dst3, w3, src2, dst2, w2, remap, K2, E2);
    build_graph_norm(isq3, dst3, w3, E2, K2);
    G g3{src3, dst3, w3, isq3, E2, K2};

    copyF(res2, x2, (size_t)K1 * H);
    copyF(res3, x3, (size_t)K2 * H);

    const long long NH = (long long)N * H, K1H = (long long)K1 * H, K2H = (long long)K2 * H;

    // helper: cross-scale pool/unpool
    auto graph_pool = [&](float* fout, const float* x, const G& g, const float* W,
                          const float* b, const float* scores, const int* sel, int k,
                          float* ta, float* tb) {
        gcn_to(tb, ta, x, W, b, g);
        k_gather_scale<<<gridL((long long)k * H), TPB, 0, stream>>>(fout, tb, scores, sel, (long long)k * H);
    };
    auto graph_unpool = [&](float* fout, const float* xs, const int* sel, int k,
                            const G& g, const float* W, const float* b,
                            float* ta, float* tb) {
        zerof(ta, (long long)g.n * H);
        k_scatter_rows<<<gridL((long long)k * H), TPB, 0, stream>>>(ta, xs, sel, (long long)k * H);
        gcn_to(fout, tb, ta, W, b, g);
    };

    // ---- main layer 0 ----
    gcn_to(t1b, t1a, x1, Wm + 0 * HH, bm + 0 * H, g1); relu(t1b, NH);  copyF(x1, t1b, NH);
    gcn_to(t2b, t2a, x2, Wm + 3 * HH, bm + 3 * H, g2); relu(t2b, K1H); copyF(x2, t2b, K1H);
    gcn_to(t3b, t3a, x3, Wm + 6 * HH, bm + 6 * H, g3); relu(t3b, K2H); copyF(x3, t3b, K2H);

    graph_pool  (f12,   x1, g1, Wp + 0 * HH, bp + 0 * H, scores1, sel1, K1, t1a, t1b);
    graph_unpool(s2buf, x2, sel1, K1, g1, Wu + 2 * HH, bu + 2 * H, t1a, t1c);  // f21
    graph_pool  (f23,   x2, g2, Wp + 1 * HH, bp + 1 * H, scores2, sel2, K2, t2a, t2b);
    graph_unpool(f32b,  x3, sel2, K2, g2, Wu + 3 * HH, bu + 3 * H, t2a, t2c);

    k_add_wsum<<<gridL(NH),  TPB, 0, stream>>>(x1, x1, s2buf, feat0, 1.0f, 1.0f, NH);
    k_add_wsum<<<gridL(K1H), TPB, 0, stream>>>(x2, x2, f12, f32b, 0.5f, 0.5f, K1H);
    k_add_wsum<<<gridL(K1H), TPB, 0, stream>>>(x2, x2, res2, nullptr, 1.0f, 0.0f, K1H);
    k_add_wsum<<<gridL(K2H), TPB, 0, stream>>>(x3, x3, f23, res3, 1.0f, 1.0f, K2H);

    // ---- main layer 1 ----
    gcn_to(t1b, t1a, x1, Wm + 1 * HH, bm + 1 * H, g1); relu(t1b, NH);  copyF(x1, t1b, NH);
    gcn_to(t2b, t2a, x2, Wm + 4 * HH, bm + 4 * H, g2); relu(t2b, K1H); copyF(x2, t2b, K1H);
    gcn_to(t3b, t3a, x3, Wm + 7 * HH, bm + 7 * H, g3); relu(t3b, K2H); copyF(x3, t3b, K2H);

    graph_pool  (f12,   x1, g1, Wp + 2 * HH, bp + 2 * H, scores1, sel1, K1, t1a, t1b);
    graph_unpool(s2buf, x2, sel1, K1, g1, Wu + 4 * HH, bu + 4 * H, t1a, t1c);  // f21
    graph_pool  (f23,   x2, g2, Wp + 3 * HH, bp + 3 * H, scores2, sel2, K2, t2a, t2b);
    graph_unpool(f32b,  x3, sel2, K2, g2, Wu + 5 * HH, bu + 5 * H, t2a, t2c);

    k_add_wsum<<<gridL(NH),  TPB, 0, stream>>>(x1, x1, s2buf, nullptr, 0.05f, 0.0f, NH);
    k_add_wsum<<<gridL(K1H), TPB, 0, stream>>>(x2, x2, f12, f32b, 0.025f, 0.025f, K1H);
    k_add_wsum<<<gridL(K2H), TPB, 0, stream>>>(x3, x3, f23, nullptr, 0.05f, 0.0f, K2H);

    // ---- main layer 2 ----
    gcn_to(t1b, t1a, x1, Wm + 2 * HH, bm + 2 * H, g1); relu(t1b, NH);  copyF(x1, t1b, NH);
    gcn_to(t2b, t2a, x2, Wm + 5 * HH, bm + 5 * H, g2); relu(t2b, K1H); copyF(x2, t2b, K1H);
    gcn_to(t3b, t3a, x3, Wm + 8 * HH, bm + 8 * H, g3); relu(t3b, K2H); copyF(x3, t3b, K2H);

    // s3_out = unpool(x3 -> K1, Wu1) + down2
    graph_unpool(s3buf, x3, sel2, K2, g2, Wu + 1 * HH, bu + 1 * H, t2a, t2c);
    k_add_wsum<<<gridL(K1H), TPB, 0, stream>>>(s3buf, s3buf, down2, nullptr, 1.0f, 0.0f, K1H);

    // s2_out = unpool(x2 + s3_out -> N, Wu0)
    k_add_wsum<<<gridL(K1H), TPB, 0, stream>>>(t2a, x2, s3buf, nullptr, 1.0f, 0.0f, K1H);
    zerof(t1a, NH);
    k_scatter_rows<<<gridL(K1H), TPB, 0, stream>>>(t1a, t2a, sel1, K1H);
    gcn_to(s2buf, t1c, t1a, Wu + 0 * HH, bu + 0 * H, g1);

    // agg = x1 + s2_out + down1 ; concat with feat_origin ; final gcn (K=256)
    k_add_wsum<<<gridL(NH), TPB, 0, stream>>>(t1b, x1, s2buf, down1, 1.0f, 1.0f, NH);
    k_concat<<<gridL((long long)N * 256), TPB, 0, stream>>>(catb, t1b, feat0, (long long)N * 256);
    gemm(t1a, catb, W_end, N, 256);
    fill_bias(out_x, b_end, N);
    agg(out_x, t1a, g1);
}